// TAG_LSTM_73830487818965
// MI455X (gfx1250) — compile-verified
//
#include <hip/hip_runtime.h>
#include <hip/hip_bf16.h>
#include <math.h>

#define BATCH  4
#define TSEQ   16
#define NPGC   2000
#define FIN    4
#define HID    128
#define COUT   16
#define NNODE  (TSEQ*NPGC)      // 32000
#define NEDGE  113024
#define LHS    512
#define RHS    512
#define CONVW  (COUT*NPGC)      // 32000
#define BNEPS  1e-5f

typedef float v2f __attribute__((ext_vector_type(2)));
typedef float v8f __attribute__((ext_vector_type(8)));

// ---------------------------------------------------------------------------
// Generic fp32 WMMA GEMM:  C[M,N] (+)= A[M,K] @ B
//   TRANSB=false: B is [K,N] row-major (ldb = row stride)
//   TRANSB=true : B is [N,K] row-major (PyTorch Linear weight), computes A@B^T
//   ATOMIC=false: C-tile preloaded and stored (tile uniquely owned)
//   ATOMIC=true : K split across blockIdx.y; partial tiles atomically added
// One wave32 owns one 16x16 C tile. Uses V_WMMA_F32_16X16X4_F32.
// A-frag: lane%16 = M row, K = 2*(lane/16)+vgpr.  B-frag mirrored.
// C-frag: vgpr r -> M = 16*tile + r + 8*(lane/16), N = 16*tile + lane%16.
// A-row index is CLAMPED (not guarded): padded A rows only affect C rows >= M,
// which are never stored -> no divergent branch in the inner loop.
// ---------------------------------------------------------------------------
template<bool TRANSB, bool ATOMIC>
__global__ __launch_bounds__(256) void wmma_gemm_f32(
    const float* __restrict__ A, const float* __restrict__ Bm,
    float* __restrict__ Cp,
    int M, int N, int K, int lda, int ldb, int ldc, int kLen)
{
  (void)K;
  const int lane   = threadIdx.x & 31;
  const int wave   = blockIdx.x * (blockDim.x >> 5) + (threadIdx.x >> 5);
  const int nTiles = N >> 4;
  const int mTiles = (M + 15) >> 4;
  if (wave >= mTiles * nTiles) return;
  const int mTile = wave / nTiles;
  const int nTile = wave - mTile * nTiles;
  const int m0 = mTile << 4, n0 = nTile << 4;
  const int lm    = lane & 15;
  const int kh    = (lane >> 4) << 1;       // 0 or 2
  const int col   = n0 + lm;
  const int mBase = m0 + ((lane >> 4) << 3);
  const int k0    = blockIdx.y * kLen;
  const int kEnd  = k0 + kLen;
  const bool fullTile = (mBase + 8 <= M);

  v8f acc = {0.f,0.f,0.f,0.f,0.f,0.f,0.f,0.f};
  if (!ATOMIC) {
    const float* cp = Cp + (long)mBase * ldc + col;
    if (fullTile) {
#pragma unroll
      for (int r = 0; r < 8; ++r) { acc[r] = *cp; cp += ldc; }
    } else {
#pragma unroll
      for (int r = 0; r < 8; ++r) { if (mBase + r < M) acc[r] = *cp; cp += ldc; }
    }
  }

  const int am = (m0 + lm < M) ? (m0 + lm) : (M - 1);   // clamp, no guard
  const float* arow = A + (long)am * lda;
  const float* brow = TRANSB ? (Bm + (long)col * ldb) : Bm;

  int k = k0;
  // 4x-unrolled main loop (16 K per trip)
  for (; k + 16 <= kEnd; k += 16) {
#pragma unroll
    for (int u = 0; u < 4; ++u) {
      const int kk = k + 4 * u + kh;
      v2f a, b;
      a[0] = arow[kk];
      a[1] = arow[kk + 1];
      if (TRANSB) {
        b[0] = brow[kk];
        b[1] = brow[kk + 1];
      } else {
        b[0] = Bm[(long)kk * ldb + col];
        b[1] = Bm[(long)(kk + 1) * ldb + col];
      }
      acc = __builtin_amdgcn_wmma_f32_16x16x4_f32(false, a, false, b, (short)0, acc,
                                                  false, false);
    }
    if (TRANSB) __builtin_prefetch(brow + k + 256, 0, 1);  // stream weights ahead
  }
  // remainder (covers K=4 layer-0 case)
  for (; k < kEnd; k += 4) {
    const int kk = k + kh;
    v2f a, b;
    a[0] = arow[kk];
    a[1] = arow[kk + 1];
    if (TRANSB) {
      b[0] = brow[kk];
      b[1] = brow[kk + 1];
    } else {
      b[0] = Bm[(long)kk * ldb + col];
      b[1] = Bm[(long)(kk + 1) * ldb + col];
    }
    acc = __builtin_amdgcn_wmma_f32_16x16x4_f32(false, a, false, b, (short)0, acc,
                                                false, false);
  }

  float* cp = Cp + (long)mBase * ldc + col;
  if (fullTile) {
#pragma unroll
    for (int r = 0; r < 8; ++r) {
      if (ATOMIC) atomicAdd(cp, acc[r]); else *cp = acc[r];
      cp += ldc;
    }
  } else {
#pragma unroll
    for (int r = 0; r < 8; ++r) {
      if (mBase + r < M) { if (ATOMIC) atomicAdd(cp, acc[r]); else *cp = acc[r]; }
      cp += ldc;
    }
  }
}

// ---------------------------------------------------------------------------
// Elementwise / graph utility kernels
// ---------------------------------------------------------------------------
__global__ void fill_kernel(float* p, float v, long n) {
  long i = (long)blockIdx.x * blockDim.x + threadIdx.x;
  if (i < n) p[i] = v;
}
__global__ void copy_kernel(float* d, const float* s, long n) {
  long i = (long)blockIdx.x * blockDim.x + threadIdx.x;
  if (i < n) d[i] = s[i];
}
__global__ void relu_kernel(float* d, const float* s, long n) {
  long i = (long)blockIdx.x * blockDim.x + threadIdx.x;
  if (i < n) d[i] = fmaxf(s[i], 0.f);
}
__global__ void bias_rows_kernel(float* C, const float* b, int M, int N, int ldc) {
  long i = (long)blockIdx.x * blockDim.x + threadIdx.x;
  if (i >= (long)M * N) return;
  int m = (int)(i / N), n = (int)(i - (long)m * N);
  C[(long)m * ldc + n] = b[n];
}
__global__ void bias2_rows_kernel(float* C, const float* b1, const float* b2,
                                  int M, int N, int ldc) {
  long i = (long)blockIdx.x * blockDim.x + threadIdx.x;
  if (i >= (long)M * N) return;
  int m = (int)(i / N), n = (int)(i - (long)m * N);
  C[(long)m * ldc + n] = b1[n] + b2[n];
}
__global__ void gather_rows_kernel(float* dst, const float* src, long rowStride,
                                   int M, int N) {
  long i = (long)blockIdx.x * blockDim.x + threadIdx.x;
  if (i >= (long)M * N) return;
  int m = (int)(i / N), n = (int)(i - (long)m * N);
  dst[i] = src[(long)m * rowStride + n];
}

__global__ void edge_w_deg_kernel(const float* ea, const int* dst, float* wbuf,
                                  float* deg, int E) {
  int e = blockIdx.x * blockDim.x + threadIdx.x;
  if (e >= E) return;
  float a0 = ea[2 * e], a1 = ea[2 * e + 1];
  float w = sqrtf(a0 * a0 + a1 * a1);
  wbuf[e] = w;
  atomicAdd(&deg[dst[e]], w);
}
__global__ void deg_to_dis_kernel(float* deg, int N) {
  int n = blockIdx.x * blockDim.x + threadIdx.x;
  if (n >= N) return;
  float d = deg[n];
  deg[n] = (d > 0.f) ? rsqrtf(d) : 0.f;
}
__global__ void edge_norm_kernel(const float* wbuf, const float* dis, const int* src,
                                 const int* dst, float* nrm, int E) {
  int e = blockIdx.x * blockDim.x + threadIdx.x;
  if (e >= E) return;
  nrm[e] = dis[src[e]] * wbuf[e] * dis[dst[e]];
}
// hnext[dst] += norm * hcur[src] ; grid.y covers feature quads (C/4)
__global__ void scatter_add_kernel(const float* hsrc, float* hdst, const int* src,
                                   const int* dst, const float* nrm, int E, int C) {
  int e = blockIdx.x * blockDim.x + threadIdx.x;
  if (e >= E) return;
  int q = blockIdx.y << 2;
  float nm = nrm[e];
  const float4 hv = *(const float4*)(hsrc + (long)src[e] * C + q);  // b128 load
  float* dp = hdst + (long)dst[e] * C + q;
  atomicAdd(dp + 0, nm * hv.x);
  atomicAdd(dp + 1, nm * hv.y);
  atomicAdd(dp + 2, nm * hv.z);
  atomicAdd(dp + 3, nm * hv.w);
}

// out = relu(BN(acc + bias)); optional skip: out = (out + skip)*0.5
__global__ void tag_epilogue_kernel(const float* acc, const float* bias,
                                    const float* g, const float* be,
                                    const float* mm, const float* vv,
                                    const float* skip, float* out, long total, int C) {
  long i = (long)blockIdx.x * blockDim.x + threadIdx.x;
  if (i >= total) return;
  int c = (int)(i % C);
  float val = acc[i] + bias[c];
  val = (val - mm[c]) * rsqrtf(vv[c] + BNEPS) * g[c] + be[c];
  val = fmaxf(val, 0.f);
  if (skip) val = (val + skip[i]) * 0.5f;
  out[i] = val;
}
// last TAG layer: BN+ReLU then reshape [N,16] -> [T, 32000] into conv (per batch)
__global__ void tag_epilogue2_kernel(const float* acc, const float* bias,
                                     const float* g, const float* be,
                                     const float* mm, const float* vv,
                                     float* conv, int Nn, int C) {
  long i = (long)blockIdx.x * blockDim.x + threadIdx.x;
  if (i >= (long)Nn * C) return;
  int n = (int)(i / C), c = (int)(i - (long)n * C);
  float val = acc[i] + bias[c];
  val = (val - mm[c]) * rsqrtf(vv[c] + BNEPS) * g[c] + be[c];
  val = fmaxf(val, 0.f);
  int t = n / NPGC, p = n - t * NPGC;
  conv[(long)t * CONVW + p * C + c] = val;
}

__device__ __forceinline__ float sigf(float x) { return 1.f / (1.f + expf(-x)); }

// z = [B,4*LH] gates (i,f,g,o); updates h,c; optionally writes h into hseq rows
__global__ void lstm_cell_kernel(const float* z, float* h, float* c,
                                 float* hseq, long hseqStride, int Bsz, int L) {
  int i = blockIdx.x * blockDim.x + threadIdx.x;
  if (i >= Bsz * L) return;
  int b = i / L, j = i - b * L;
  const float* zr = z + (long)b * 4 * L;
  float ig = sigf(zr[j]);
  float fg = sigf(zr[L + j]);
  float gg = tanhf(zr[2 * L + j]);
  float og = sigf(zr[3 * L + j]);
  float cn = fg * c[i] + ig * gg;
  c[i] = cn;
  float hn = og * tanhf(cn);
  h[i] = hn;
  if (hseq) hseq[(long)b * hseqStride + j] = hn;
}

// ---------------------------------------------------------------------------
extern "C" void kernel_launch(void* const* d_in, const int* in_sizes, int n_in,
                              void* d_out, int out_size, void* d_ws, size_t ws_size,
                              hipStream_t stream) {
  (void)in_sizes; (void)n_in; (void)out_size; (void)ws_size;
  const float* x    = (const float*)d_in[0];
  const int*   ei   = (const int*)  d_in[1];
  const float* ea   = (const float*)d_in[2];
  const float* tW0  = (const float*)d_in[3];  const float* tb0 = (const float*)d_in[4];
  const float* tW1  = (const float*)d_in[5];  const float* tb1 = (const float*)d_in[6];
  const float* tW2  = (const float*)d_in[7];  const float* tb2 = (const float*)d_in[8];
  const float* g0   = (const float*)d_in[9];  const float* be0 = (const float*)d_in[10];
  const float* m0   = (const float*)d_in[11]; const float* v0  = (const float*)d_in[12];
  const float* g1   = (const float*)d_in[13]; const float* be1 = (const float*)d_in[14];
  const float* m1   = (const float*)d_in[15]; const float* v1  = (const float*)d_in[16];
  const float* g2   = (const float*)d_in[17]; const float* be2 = (const float*)d_in[18];
  const float* m2   = (const float*)d_in[19]; const float* v2  = (const float*)d_in[20];
  const float* Wih0 = (const float*)d_in[21]; const float* Whh0 = (const float*)d_in[22];
  const float* bih0 = (const float*)d_in[23]; const float* bhh0 = (const float*)d_in[24];
  const float* Wih1 = (const float*)d_in[25]; const float* Whh1 = (const float*)d_in[26];
  const float* bih1 = (const float*)d_in[27]; const float* bhh1 = (const float*)d_in[28];
  const float* rW0  = (const float*)d_in[29]; const float* rb0 = (const float*)d_in[30];
  const float* rW1  = (const float*)d_in[31]; const float* rb1 = (const float*)d_in[32];
  const float* f1W  = (const float*)d_in[33]; const float* f1b = (const float*)d_in[34];
  const float* f2W  = (const float*)d_in[35]; const float* f2b = (const float*)d_in[36];
  const float* eW   = (const float*)d_in[37]; const float* eb  = (const float*)d_in[38];
  float* out = (float*)d_out;

  // workspace layout (floats); per-batch graph buffers reused (stay L2-resident)
  float* ws = (float*)d_ws;
  auto wsalloc = [&](size_t n) { float* p = ws; ws += n; return p; };
  float* deg    = wsalloc(NNODE);
  float* wbuf   = wsalloc(NEDGE);
  float* nrm    = wsalloc(NEDGE);
  float* hA     = wsalloc((size_t)NNODE * HID);
  float* hB     = wsalloc((size_t)NNODE * HID);
  float* acc    = wsalloc((size_t)NNODE * HID);
  float* l0out  = wsalloc((size_t)NNODE * HID);
  float* l1out  = wsalloc((size_t)NNODE * HID);
  float* conv   = wsalloc((size_t)BATCH * TSEQ * CONVW);
  float* xproj0 = wsalloc((size_t)BATCH * TSEQ * 4 * LHS);
  float* hseq0  = wsalloc((size_t)BATCH * TSEQ * LHS);
  float* xproj1 = wsalloc((size_t)BATCH * TSEQ * 4 * LHS);
  float* zbuf   = wsalloc((size_t)BATCH * 4 * LHS);
  float* h0     = wsalloc((size_t)BATCH * LHS);
  float* c0     = wsalloc((size_t)BATCH * LHS);
  float* h1     = wsalloc((size_t)BATCH * LHS);
  float* c1     = wsalloc((size_t)BATCH * LHS);
  float* tA     = wsalloc((size_t)BATCH * RHS);
  float* tB     = wsalloc((size_t)BATCH * RHS);

  auto fill = [&](float* p, float v, long n) {
    fill_kernel<<<dim3((unsigned)((n + 255) / 256)), dim3(256), 0, stream>>>(p, v, n);
  };
  auto copyb = [&](float* d, const float* s, long n) {
    copy_kernel<<<dim3((unsigned)((n + 255) / 256)), dim3(256), 0, stream>>>(d, s, n);
  };
  auto gemm = [&](const float* A, const float* Bm, float* Cp, int M, int N, int K,
                  int lda, int ldb, int ldc, bool transB, bool atomic, int kChunks) {
    int tiles = ((M + 15) / 16) * (N / 16);
    dim3 grid((unsigned)((tiles + 7) / 8), (unsigned)kChunks, 1);
    dim3 blk(256, 1, 1);
    int kLen = K / kChunks;
    if (transB) {
      if (atomic) wmma_gemm_f32<true,  true ><<<grid, blk, 0, stream>>>(A, Bm, Cp, M, N, K, lda, ldb, ldc, kLen);
      else        wmma_gemm_f32<true,  false><<<grid, blk, 0, stream>>>(A, Bm, Cp, M, N, K, lda, ldb, ldc, kLen);
    } else {
      if (atomic) wmma_gemm_f32<false, true ><<<grid, blk, 0, stream>>>(A, Bm, Cp, M, N, K, lda, ldb, ldc, kLen);
      else        wmma_gemm_f32<false, false><<<grid, blk, 0, stream>>>(A, Bm, Cp, M, N, K, lda, ldb, ldc, kLen);
    }
  };

  const unsigned EB = (NEDGE + 255) / 256;

  // ============================ Graph part (per batch) =====================
  for (int b = 0; b < BATCH; ++b) {
    const float* xb   = x  + (size_t)b * NNODE * FIN;
    const int*   srcb = ei + (size_t)b * 2 * NEDGE;
    const int*   dstb = srcb + NEDGE;
    const float* eab  = ea + (size_t)b * NEDGE * 2;

    fill(deg, 0.f, NNODE);
    edge_w_deg_kernel<<<EB, 256, 0, stream>>>(eab, dstb, wbuf, deg, NEDGE);
    deg_to_dis_kernel<<<(NNODE + 255) / 256, 256, 0, stream>>>(deg, NNODE);
    edge_norm_kernel<<<EB, 256, 0, stream>>>(wbuf, deg, srcb, dstb, nrm, NEDGE);

    // ---- TAG layer 0: [N,4] -> [N,128] ----
    copyb(hA, xb, (long)NNODE * FIN);
    fill(acc, 0.f, (long)NNODE * HID);
    {
      float* cur = hA; float* nxt = hB;
      for (int hop = 0; hop <= 3; ++hop) {
        gemm(cur, tW0 + (size_t)hop * FIN * HID, acc,
             NNODE, HID, FIN, FIN, HID, HID, false, false, 1);
        if (hop < 3) {
          fill(nxt, 0.f, (long)NNODE * FIN);
          scatter_add_kernel<<<dim3(EB, FIN / 4), 256, 0, stream>>>(
              cur, nxt, srcb, dstb, nrm, NEDGE, FIN);
          float* t = cur; cur = nxt; nxt = t;
        }
      }
    }
    tag_epilogue_kernel<<<((long)NNODE * HID + 255) / 256, 256, 0, stream>>>(
        acc, tb0, g0, be0, m0, v0, nullptr, l0out, (long)NNODE * HID, HID);

    // ---- TAG layer 1: [N,128] -> [N,128], skip-add with l0out ----
    copyb(hA, l0out, (long)NNODE * HID);
    fill(acc, 0.f, (long)NNODE * HID);
    {
      float* cur = hA; float* nxt = hB;
      for (int hop = 0; hop <= 3; ++hop) {
        gemm(cur, tW1 + (size_t)hop * HID * HID, acc,
             NNODE, HID, HID, HID, HID, HID, false, false, 1);
        if (hop < 3) {
          fill(nxt, 0.f, (long)NNODE * HID);
          scatter_add_kernel<<<dim3(EB, HID / 4), 256, 0, stream>>>(
              cur, nxt, srcb, dstb, nrm, NEDGE, HID);
          float* t = cur; cur = nxt; nxt = t;
        }
      }
    }
    tag_epilogue_kernel<<<((long)NNODE * HID + 255) / 256, 256, 0, stream>>>(
        acc, tb1, g1, be1, m1, v1, l0out, l1out, (long)NNODE * HID, HID);

    // ---- TAG layer 2: [N,128] -> [N,16] -> conv[b] reshaped [T, 32000] ----
    copyb(hA, l1out, (long)NNODE * HID);
    fill(acc, 0.f, (long)NNODE * COUT);
    {
      float* cur = hA; float* nxt = hB;
      for (int hop = 0; hop <= 3; ++hop) {
        gemm(cur, tW2 + (size_t)hop * HID * COUT, acc,
             NNODE, COUT, HID, HID, COUT, COUT, false, false, 1);
        if (hop < 3) {
          fill(nxt, 0.f, (long)NNODE * HID);
          scatter_add_kernel<<<dim3(EB, HID / 4), 256, 0, stream>>>(
              cur, nxt, srcb, dstb, nrm, NEDGE, HID);
          float* t = cur; cur = nxt; nxt = t;
        }
      }
    }
    tag_epilogue2_kernel<<<((long)NNODE * COUT + 255) / 256, 256, 0, stream>>>(
        acc, tb2, g2, be2, m2, v2, conv + (size_t)b * TSEQ * CONVW, NNODE, COUT);
  }

  // ============================ LSTM layer 0 ===============================
  // xproj0[b*T+t,:] = bih0+bhh0 + conv[b,t,:] @ Wih0^T  (262MB weight stream, K-split)
  bias2_rows_kernel<<<(BATCH * TSEQ * 4 * LHS + 255) / 256, 256, 0, stream>>>(
      xproj0, bih0, bhh0, BATCH * TSEQ, 4 * LHS, 4 * LHS);
  gemm(conv, Wih0, xproj0, BATCH * TSEQ, 4 * LHS, CONVW,
       CONVW, CONVW, 4 * LHS, true, true, 16);

  fill(h0, 0.f, BATCH * LHS); fill(c0, 0.f, BATCH * LHS);
  fill(h1, 0.f, BATCH * LHS); fill(c1, 0.f, BATCH * LHS);

  for (int t = 0; t < TSEQ; ++t) {
    gather_rows_kernel<<<(BATCH * 4 * LHS + 255) / 256, 256, 0, stream>>>(
        zbuf, xproj0 + (size_t)t * 4 * LHS, (long)TSEQ * 4 * LHS, BATCH, 4 * LHS);
    gemm(h0, Whh0, zbuf, BATCH, 4 * LHS, LHS, LHS, LHS, 4 * LHS, true, false, 1);
    lstm_cell_kernel<<<(BATCH * LHS + 255) / 256, 256, 0, stream>>>(
        zbuf, h0, c0, hseq0 + (size_t)t * LHS, (long)TSEQ * LHS, BATCH, LHS);
  }

  // ============================ LSTM layer 1 ===============================
  bias2_rows_kernel<<<(BATCH * TSEQ * 4 * LHS + 255) / 256, 256, 0, stream>>>(
      xproj1, bih1, bhh1, BATCH * TSEQ, 4 * LHS, 4 * LHS);
  gemm(hseq0, Wih1, xproj1, BATCH * TSEQ, 4 * LHS, LHS, LHS, LHS, 4 * LHS, true, false, 1);

  for (int t = 0; t < TSEQ; ++t) {
    gather_rows_kernel<<<(BATCH * 4 * LHS + 255) / 256, 256, 0, stream>>>(
        zbuf, xproj1 + (size_t)t * 4 * LHS, (long)TSEQ * 4 * LHS, BATCH, 4 * LHS);
    gemm(h1, Whh1, zbuf, BATCH, 4 * LHS, LHS, LHS, LHS, 4 * LHS, true, false, 1);
    lstm_cell_kernel<<<(BATCH * LHS + 255) / 256, 256, 0, stream>>>(
        zbuf, h1, c1, nullptr, 0, BATCH, LHS);
  }

  // ============================ Heads ======================================
  relu_kernel<<<(BATCH * LHS + 255) / 256, 256, 0, stream>>>(tA, h1, BATCH * LHS);

  bias_rows_kernel<<<(BATCH * RHS + 255) / 256, 256, 0, stream>>>(tB, rb0, BATCH, RHS, RHS);
  gemm(tA, rW0, tB, BATCH, RHS, LHS, LHS, LHS, RHS, true, false, 1);
  relu_kernel<<<(BATCH * RHS + 255) / 256, 256, 0, stream>>>(tB, tB, BATCH * RHS);

  bias_rows_kernel<<<(BATCH * RHS + 255) / 256, 256, 0, stream>>>(tA, rb1, BATCH, RHS, RHS);
  gemm(tB, rW1, tA, BATCH, RHS, RHS, RHS, RHS, RHS, true, false, 1);

  // f1/f2 interleaved as [B,2,2000] at out[0..16000); eo as [B,2,7064] after
  bias_rows_kernel<<<(BATCH * 2000 + 255) / 256, 256, 0, stream>>>(out, f1b, BATCH, 2000, 4000);
  gemm(tA, f1W, out, BATCH, 2000, RHS, RHS, RHS, 4000, true, false, 1);

  bias_rows_kernel<<<(BATCH * 2000 + 255) / 256, 256, 0, stream>>>(out + 2000, f2b, BATCH, 2000, 4000);
  gemm(tA, f2W, out + 2000, BATCH, 2000, RHS, RHS, RHS, 4000, true, false, 1);

  bias_rows_kernel<<<(BATCH * 14128 + 255) / 256, 256, 0, stream>>>(out + 16000, eb, BATCH, 14128, 14128);
  gemm(tA, eW, out + 16000, BATCH, 14128, RHS, RHS, RHS, 14128, true, false, 1);
}